// GAT_43078521979008
// MI455X (gfx1250) — compile-verified
//
#include <hip/hip_runtime.h>

// ---------------------------------------------------------------------------
// GAT (3-layer) for MI455X / gfx1250.  GEMMs via V_WMMA_F32_16X16X4_F32 (exact
// f32), edge softmax + aggregation via native f32 global atomics.  wave32.
// ---------------------------------------------------------------------------

typedef float v2f __attribute__((ext_vector_type(2)));
typedef float v8f __attribute__((ext_vector_type(8)));

#if defined(__has_builtin)
#if __has_builtin(__builtin_amdgcn_wmma_f32_16x16x4_f32)
#define HAS_WMMA_F32X4 1
#endif
#endif

#define NEG_SLOPE 0.2f

// Monotonic uint encoding of float for atomicMax-based segment-max.
__device__ __forceinline__ unsigned fkey(float f) {
  unsigned u = __float_as_uint(f);
  return (u & 0x80000000u) ? ~u : (u | 0x80000000u);
}
__device__ __forceinline__ float fdec(unsigned u) {
  unsigned b = (u & 0x80000000u) ? (u ^ 0x80000000u) : ~u;
  return __uint_as_float(b);
}

// ---------------------------------------------------------------------------
// C[nRows x M] = A[nRows x K] * B[K x M], row-major f32.  K, M compile-time;
// nRows must be a multiple of 16 (N = 100000 = 6250*16).  One wave per 16x16
// output tile; fully unrolled K loop -> 32 back-to-back WMMAs with batched
// b64/b32 loads (no exec churn, immediate address offsets).
// ISA layouts (7.12.2): A lane m=lane&15 holds K=kh,kh+1 (kh = 2*(lane>=16));
// B lane's column = lane&15, rows K=kh,kh+1; C/D vgpr j -> row rb+j where
// rb = row0 + 8*(lane>=16), col = col0 + (lane&15).
// ---------------------------------------------------------------------------
template <int K, int M>
__global__ void __launch_bounds__(256) gemm_wmma_f32(const float* __restrict__ A,
                                                     const float* __restrict__ B,
                                                     float* __restrict__ C, int nRows) {
  constexpr int TPR = M / 16;                  // column tiles per row tile
  const int lane = threadIdx.x & 31;
  const int wv   = threadIdx.x >> 5;
  const int tile = blockIdx.x * 8 + wv;
  const int nTiles = (nRows >> 4) * TPR;
  if (tile >= nTiles) return;                  // wave-uniform: EXEC all-1s for WMMA
  const int tr   = tile / TPR;
  const int tc   = tile % TPR;
  const int row0 = tr << 4, col0 = tc << 4;
  const int m    = lane & 15;
  const int kh   = (lane >> 4) << 1;           // 0 or 2
  const float* Ap = A + (size_t)(row0 + m) * K + kh;   // 8B aligned (kh even)
  const float* Bp = B + (size_t)kh * M + col0 + m;
  const int rb   = row0 + ((lane >> 4) << 3);
  const int col  = col0 + m;

  v8f c = {0.f, 0.f, 0.f, 0.f, 0.f, 0.f, 0.f, 0.f};
  #pragma unroll
  for (int k0 = 0; k0 < K; k0 += 4) {
#ifdef HAS_WMMA_F32X4
    v2f a = *reinterpret_cast<const v2f*>(Ap + k0);    // global_load_b64
    v2f b;
    b[0] = Bp[k0 * M];                                 // immediate offsets
    b[1] = Bp[k0 * M + M];
    c = __builtin_amdgcn_wmma_f32_16x16x4_f32(false, a, false, b,
                                              (short)0, c, false, false);
#else
    #pragma unroll
    for (int q = 0; q < 4; ++q) {
      float bk = B[(size_t)(k0 + q) * M + col];
      #pragma unroll
      for (int j = 0; j < 8; ++j)
        c[j] += A[(size_t)(rb + j) * K + (k0 + q)] * bk;
    }
#endif
  }
  float* Cp = C + (size_t)rb * M + col;
  #pragma unroll
  for (int j = 0; j < 8; ++j) Cp[j * M] = c[j];
}

// el/er[n*Hh+h] = sum_d feat[n][h][d] * {al,ar}[h][d]
template <int Dd, int Hh>
__global__ void attn_scores(const float* __restrict__ feat,
                            const float* __restrict__ al, const float* __restrict__ ar,
                            float* __restrict__ el, float* __restrict__ er, int nH) {
  int idx = blockIdx.x * blockDim.x + threadIdx.x;
  if (idx >= nH) return;
  const int h = idx % Hh;
  const float4* f4 = reinterpret_cast<const float4*>(feat + (size_t)idx * Dd);
  const float4* a4 = reinterpret_cast<const float4*>(al + h * Dd);
  const float4* b4 = reinterpret_cast<const float4*>(ar + h * Dd);
  float sl = 0.f, sr = 0.f;
  #pragma unroll
  for (int i = 0; i < Dd / 4; ++i) {
    float4 f = f4[i], a = a4[i], b = b4[i];
    sl += f.x * a.x + f.y * a.y + f.z * a.z + f.w * a.w;
    sr += f.x * b.x + f.y * b.y + f.z * b.z + f.w * b.w;
  }
  el[idx] = sl;
  er[idx] = sr;
}

// pass1: e = leaky_relu(el[src]+er[dst]); segment-max via uint atomicMax
template <int Hh>
__global__ void edge_pass1(const int* __restrict__ src, const int* __restrict__ dst,
                           const float* __restrict__ el, const float* __restrict__ er,
                           float* __restrict__ ebuf, unsigned* __restrict__ emax, int EH) {
  int idx = blockIdx.x * blockDim.x + threadIdx.x;
  if (idx >= EH) return;
  const int e = idx / Hh, h = idx - e * Hh;
  const int s = src[e], d = dst[e];
  float v = el[s * Hh + h] + er[d * Hh + h];
  v = v > 0.f ? v : NEG_SLOPE * v;
  ebuf[idx] = v;
  atomicMax(emax + d * Hh + h, fkey(v));
}

// pass2: ex = exp(e - emax[dst]); den[dst] += ex
template <int Hh>
__global__ void edge_pass2(const int* __restrict__ dst, float* __restrict__ ebuf,
                           const unsigned* __restrict__ emax, float* __restrict__ den, int EH) {
  int idx = blockIdx.x * blockDim.x + threadIdx.x;
  if (idx >= EH) return;
  const int e = idx / Hh, h = idx - e * Hh;
  const int d = dst[e];
  float x = __expf(ebuf[idx] - fdec(emax[d * Hh + h]));
  ebuf[idx] = x;
  atomicAdd(den + d * Hh + h, x);
}

// pass3 (dominant traffic): agg[dst][f] += feat[src][f] * (ex / den[dst])
// F threads per edge -> coalesced 4B*F gather + native f32 atomic scatter.
template <int F, int Hh, int Dd>
__global__ void edge_pass3(const int* __restrict__ src, const int* __restrict__ dst,
                           const float* __restrict__ feat, const float* __restrict__ ex,
                           const float* __restrict__ den, float* __restrict__ agg, int E) {
  const int slots = blockDim.x / F;
  const int slot  = threadIdx.x / F;
  const int f     = threadIdx.x - slot * F;
  const int h     = f / Dd;
  for (int e = blockIdx.x * slots + slot; e < E; e += gridDim.x * slots) {
    const int s = src[e], d = dst[e];
    const float a = ex[e * Hh + h] / den[d * Hh + h];
    atomicAdd(agg + (size_t)d * F + f, feat[(size_t)s * F + f] * a);
  }
}

// relu(elu(x)) == relu(x); fused epilogues.
__global__ void relu_copy(const float* __restrict__ agg, float* __restrict__ h, int n) {
  int i = blockIdx.x * blockDim.x + threadIdx.x;
  if (i < n) h[i] = fmaxf(agg[i], 0.f);
}
__global__ void relu_add(const float* __restrict__ agg, float* __restrict__ h, int n) {
  int i = blockIdx.x * blockDim.x + threadIdx.x;
  if (i < n) h[i] = fmaxf(agg[i] + h[i], 0.f);
}
__global__ void add_out(const float* __restrict__ agg, const float* __restrict__ res,
                        float* __restrict__ out, int n) {
  int i = blockIdx.x * blockDim.x + threadIdx.x;
  if (i < n) out[i] = agg[i] + res[i];
}

// ---------------------------------------------------------------------------
extern "C" void kernel_launch(void* const* d_in, const int* in_sizes, int n_in,
                              void* d_out, int out_size, void* d_ws, size_t ws_size,
                              hipStream_t stream) {
  const float* inputs = (const float*)d_in[0];
  const int*   src    = (const int*)d_in[1];
  const int*   dst    = (const int*)d_in[2];
  const float* W0  = (const float*)d_in[3];
  const float* al0 = (const float*)d_in[4];
  const float* ar0 = (const float*)d_in[5];
  const float* W1  = (const float*)d_in[6];
  const float* al1 = (const float*)d_in[7];
  const float* ar1 = (const float*)d_in[8];
  const float* W2  = (const float*)d_in[9];
  const float* al2 = (const float*)d_in[10];
  const float* ar2 = (const float*)d_in[11];
  const float* rW2 = (const float*)d_in[12];

  const int N = in_sizes[0] / 128;   // IN = 128, N divisible by 16
  const int E = in_sizes[1];
  const int Hc = 4, F = 128, Cc = 16;

  float* ws = (float*)d_ws;
  const size_t nf = (size_t)N * F;
  float*    feat = ws;                 // N*128 (layer2: feat2 in [0,N*16))
  float*    agg  = ws + nf;            // N*128
  float*    hbuf = ws + 2 * nf;        // N*128
  float*    ebuf = ws + 3 * nf;        // E*4
  float*    el   = ebuf + (size_t)E * Hc;       // N*4
  float*    er   = el + (size_t)N * Hc;         // N*4
  unsigned* emax = (unsigned*)(er + (size_t)N * Hc);   // N*4
  float*    den  = (float*)(emax + (size_t)N * Hc);    // N*4
  float*    res2 = feat + (size_t)N * Cc;       // N*16, packed inside feat region

  const int BLK = 256;
  const int tilesN = N / 16;

  // ------------------------- layer 0 (no residual) -------------------------
  gemm_wmma_f32<128, 128><<<(tilesN * 8 + 7) / 8, BLK, 0, stream>>>(inputs, W0, feat, N);
  attn_scores<32, 4><<<(N * Hc + BLK - 1) / BLK, BLK, 0, stream>>>(feat, al0, ar0, el, er, N * Hc);
  hipMemsetAsync(emax, 0, (size_t)N * Hc * 4, stream);
  hipMemsetAsync(den,  0, (size_t)N * Hc * 4, stream);
  hipMemsetAsync(agg,  0, nf * 4, stream);
  edge_pass1<4><<<(E * Hc + BLK - 1) / BLK, BLK, 0, stream>>>(src, dst, el, er, ebuf, emax, E * Hc);
  edge_pass2<4><<<(E * Hc + BLK - 1) / BLK, BLK, 0, stream>>>(dst, ebuf, emax, den, E * Hc);
  edge_pass3<128, 4, 32><<<(E + 1) / 2, BLK, 0, stream>>>(src, dst, feat, ebuf, den, agg, E);
  relu_copy<<<((int)nf + BLK - 1) / BLK, BLK, 0, stream>>>(agg, hbuf, (int)nf);

  // --------------------- layer 1 (identity residual) -----------------------
  gemm_wmma_f32<128, 128><<<(tilesN * 8 + 7) / 8, BLK, 0, stream>>>(hbuf, W1, feat, N);
  attn_scores<32, 4><<<(N * Hc + BLK - 1) / BLK, BLK, 0, stream>>>(feat, al1, ar1, el, er, N * Hc);
  hipMemsetAsync(emax, 0, (size_t)N * Hc * 4, stream);
  hipMemsetAsync(den,  0, (size_t)N * Hc * 4, stream);
  hipMemsetAsync(agg,  0, nf * 4, stream);
  edge_pass1<4><<<(E * Hc + BLK - 1) / BLK, BLK, 0, stream>>>(src, dst, el, er, ebuf, emax, E * Hc);
  edge_pass2<4><<<(E * Hc + BLK - 1) / BLK, BLK, 0, stream>>>(dst, ebuf, emax, den, E * Hc);
  edge_pass3<128, 4, 32><<<(E + 1) / 2, BLK, 0, stream>>>(src, dst, feat, ebuf, den, agg, E);
  relu_add<<<((int)nf + BLK - 1) / BLK, BLK, 0, stream>>>(agg, hbuf, (int)nf);

  // ------------- layer 2 (1 head, C=16, linear residual proj) --------------
  gemm_wmma_f32<128, 16><<<(tilesN + 7) / 8, BLK, 0, stream>>>(hbuf, W2, feat, N);
  gemm_wmma_f32<128, 16><<<(tilesN + 7) / 8, BLK, 0, stream>>>(hbuf, rW2, res2, N);
  attn_scores<16, 1><<<(N + BLK - 1) / BLK, BLK, 0, stream>>>(feat, al2, ar2, el, er, N);
  hipMemsetAsync(emax, 0, (size_t)N * 4, stream);
  hipMemsetAsync(den,  0, (size_t)N * 4, stream);
  hipMemsetAsync(agg,  0, (size_t)N * Cc * 4, stream);
  edge_pass1<1><<<(E + BLK - 1) / BLK, BLK, 0, stream>>>(src, dst, el, er, ebuf, emax, E);
  edge_pass2<1><<<(E + BLK - 1) / BLK, BLK, 0, stream>>>(dst, ebuf, emax, den, E);
  edge_pass3<16, 1, 16><<<(E + 15) / 16, BLK, 0, stream>>>(src, dst, feat, ebuf, den, agg, E);
  add_out<<<(N * Cc + BLK - 1) / BLK, BLK, 0, stream>>>(agg, res2, (float*)d_out, N * Cc);
}